// KNNGraph_51384988729794
// MI455X (gfx1250) — compile-verified
//
#include <hip/hip_runtime.h>

// kNN graph: N_SETS=8, N_POINTS=4096, D_FEAT=16, K=16
// dist = ||xi||^2 + ||xj||^2 - 2 * (X X^T)  via V_WMMA_F32_16X16X4_F32

#define NSETS 8
#define NPTS  4096
#define DF    16
#define KSEL  16
#define ROWS_PER_BLOCK 256
#define COLS_PER_ITER  16
#define ASTRIDE 17   // LDS stride (floats) for cached row features
#define DSTRIDE 18   // LDS stride (floats) for distance tile (bank-conflict free)

typedef float v2f __attribute__((ext_vector_type(2)));
typedef float v8f __attribute__((ext_vector_type(8)));

__global__ __launch_bounds__(256)
void KNNGraph_51384988729794_kernel(const float* __restrict__ x,
                                    float* __restrict__ out)
{
    __shared__ float xa[ROWS_PER_BLOCK * ASTRIDE];     // row features
    __shared__ float x2row[ROWS_PER_BLOCK];            // row squared norms
    __shared__ float dtile[ROWS_PER_BLOCK * DSTRIDE];  // 256 x 16 distance tile
    __shared__ float x2col[COLS_PER_ITER];             // column squared norms

    const int tid  = threadIdx.x;
    const int lane = tid & 31;
    const int wave = tid >> 5;
    const int set  = blockIdx.x >> 4;        // 16 row-blocks per set
    const int rb   = blockIdx.x & 15;
    const int row0 = rb * ROWS_PER_BLOCK;    // row base within the set

    const float* __restrict__ xs = x + (size_t)set * NPTS * DF;

    // ---- load this block's 256 rows + norms into LDS ----
    {
        const float* rp = xs + (size_t)(row0 + tid) * DF;
        float s = 0.0f;
        #pragma unroll
        for (int f = 0; f < DF; ++f) {
            float v = rp[f];
            xa[tid * ASTRIDE + f] = v;
            s += v * v;
        }
        x2row[tid] = s;
    }
    __syncthreads();

    // ---- per-thread top-k state (sorted ascending, [K-1] = current max) ----
    float bd[KSEL];
    int   bi[KSEL];
    #pragma unroll
    for (int i = 0; i < KSEL; ++i) { bd[i] = 3.0e38f; bi[i] = 0; }

    const int half  = lane >> 4;      // 0: lanes 0-15, 1: lanes 16-31
    const int l16   = lane & 15;
    const int fbase = 2 * half;       // K offset within each 4-wide WMMA step

    for (int cb = 0; cb < NPTS; cb += COLS_PER_ITER) {
        // column squared norms (threads 0..15)
        if (tid < COLS_PER_ITER) {
            const float* cp = xs + (size_t)(cb + tid) * DF;
            float s = 0.0f;
            #pragma unroll
            for (int f = 0; f < DF; ++f) { float v = cp[f]; s += v * v; }
            x2col[tid] = s;
        }
        // prefetch next column tile while we work on this one
        if (cb + COLS_PER_ITER < NPTS) {
            __builtin_prefetch(xs + (size_t)(cb + COLS_PER_ITER + l16) * DF, 0, 1);
        }
        __syncthreads();

        // B fragments: B[k][n] = X[col n][feat k]; lane L holds feats
        // fbase+{0,1} of column (cb + L%16), for each K-step kk.
        v2f bfr[4];
        const float* bp = xs + (size_t)(cb + l16) * DF + fbase;
        #pragma unroll
        for (int kk = 0; kk < 4; ++kk) {
            bfr[kk] = *(const v2f*)(bp + 4 * kk);
        }

        const float x2c = x2col[l16];

        // each wave: two 16-row subtiles against this 16-column tile
        #pragma unroll
        for (int ms = 0; ms < 2; ++ms) {
            const int mrow0 = wave * 32 + ms * 16;   // local row base
            v8f acc = {};
            const float* ap = &xa[(mrow0 + l16) * ASTRIDE + fbase];
            #pragma unroll
            for (int kk = 0; kk < 4; ++kk) {
                v2f afr;
                afr.x = ap[4 * kk];
                afr.y = ap[4 * kk + 1];
                acc = __builtin_amdgcn_wmma_f32_16x16x4_f32(
                    false, afr, false, bfr[kk], (short)0, acc, false, false);
            }
            // D layout: lane L, VGPR v -> (M = v + 8*(L/16), N = L%16)
            #pragma unroll
            for (int v = 0; v < 8; ++v) {
                int r = mrow0 + v + 8 * half;
                float d = x2row[r] + x2c - 2.0f * acc[v];
                dtile[r * DSTRIDE + l16] = d;
            }
        }
        __syncthreads();

        // ---- selection: thread t owns row t; single compare in common case
        #pragma unroll
        for (int j = 0; j < COLS_PER_ITER; ++j) {
            float d  = dtile[tid * DSTRIDE + j];
            int  idx = cb + j;
            if (d < bd[KSEL - 1]) {
                bd[KSEL - 1] = d; bi[KSEL - 1] = idx;
                #pragma unroll
                for (int s = KSEL - 1; s > 0; --s) {
                    if (bd[s] < bd[s - 1]) {
                        float td = bd[s]; bd[s] = bd[s - 1]; bd[s - 1] = td;
                        int   ti = bi[s]; bi[s] = bi[s - 1]; bi[s - 1] = ti;
                    }
                }
            }
        }
        // no barrier needed here: next iteration's first barrier orders
        // these reads before the next distance-tile overwrite
    }

    // ---- write outputs: src ids, dst ids, top-k distances (as floats) ----
    const size_t E    = (size_t)NSETS * NPTS * KSEL;
    const int    grow = set * NPTS + row0 + tid;          // global node id
    const size_t base = (size_t)grow * KSEL;
    float* __restrict__ src_o  = out;
    float* __restrict__ dst_o  = out + E;
    float* __restrict__ dist_o = out + 2 * E;
    #pragma unroll
    for (int j = 0; j < KSEL; ++j) {
        src_o[base + j]  = (float)grow;
        dst_o[base + j]  = (float)(set * NPTS + bi[j]);
        dist_o[base + j] = bd[j];
    }
}

extern "C" void kernel_launch(void* const* d_in, const int* in_sizes, int n_in,
                              void* d_out, int out_size, void* d_ws, size_t ws_size,
                              hipStream_t stream)
{
    (void)in_sizes; (void)n_in; (void)out_size; (void)d_ws; (void)ws_size;
    const float* x = (const float*)d_in[0];   // (8, 4096, 16) f32; d_in[1] = k (unused)
    float* out = (float*)d_out;               // [src | dst | topk_dist], each 8*4096*16

    dim3 grid(NSETS * (NPTS / ROWS_PER_BLOCK));   // 8 * 16 = 128 blocks
    dim3 block(ROWS_PER_BLOCK);                   // 256 threads = 8 wave32
    KNNGraph_51384988729794_kernel<<<grid, block, 0, stream>>>(x, out);
}